// MODEST_31507880083952
// MI455X (gfx1250) — compile-verified
//
#include <hip/hip_runtime.h>
#include <hip/hip_bf16.h>

// ---------------------------------------------------------------------------
// Problem dimensions (fixed by setup_inputs)
// ---------------------------------------------------------------------------
#define N_ENT 20000
#define DIM   128
#define NG    2000
#define ND    500
#define DIS   256
#define R2    16
#define EK    640000
#define EG    32000
#define NP    512

typedef __attribute__((ext_vector_type(2))) float v2f;
typedef __attribute__((ext_vector_type(8))) float v8f;

// ---------------------------------------------------------------------------
// Generic f32 WMMA GEMM: C[M,N] = A[M,K] @ B  (B row-major K x N, or if
// TRANSB, B is row-major N x K and we compute A @ B^T).
//
// One wave (32 lanes) computes one 16x16 C tile via V_WMMA_F32_16X16X4_F32.
// 8 waves per 256-thread block, tiles assigned m-major so the waves of a
// block share the same B panel.  Requirements (hold at all call sites):
//   N % 16 == 0, K % 8 == 0, lda/ldb even (8-byte aligned float2 loads).
// M may be ragged: A row index is clamped (an OOB A row only feeds the
// matching OOB D row, which the epilogue store masks out) -- no branches
// in the K loop, EXEC stays all-ones at every WMMA.
// Two independent accumulator chains + unroll to overlap load latency.
// ---------------------------------------------------------------------------
template <int TRANSB>
__global__ __launch_bounds__(256) void wmma_gemm_f32(
        const float* __restrict__ A, const float* __restrict__ B,
        const float* __restrict__ bias, float* __restrict__ C,
        int M, int N, int K, int lda, int ldb, int ldc) {
    const int lane = threadIdx.x & 31;
    const int wave = threadIdx.x >> 5;
    const int tiles_m = (M + 15) >> 4;
    const int tiles_n = N >> 4;
    const int tile = blockIdx.x * 8 + wave;
    if (tile >= tiles_m * tiles_n) return;   // uniform per wave
    const int mt = tile % tiles_m;           // m-major: block's 8 waves share B
    const int nt = tile / tiles_m;

    const int half = lane >> 4;              // 0: K pair {0,1}; 1: K pair {2,3}
    const int l    = lane & 15;

    long arow = (long)mt * 16 + l;
    if (arow >= M) arow = M - 1;             // clamp; epilogue masks OOB rows
    const float* Ap = A + arow * (long)lda + half * 2;

    const long bcol = (long)nt * 16 + l;     // always < N (N % 16 == 0)
    const float* Bp = TRANSB ? (B + bcol * (long)ldb + half * 2)
                             : (B + (long)(half * 2) * ldb + bcol);

    v8f acc0 = {0.f,0.f,0.f,0.f,0.f,0.f,0.f,0.f};
    v8f acc1 = {0.f,0.f,0.f,0.f,0.f,0.f,0.f,0.f};

#pragma unroll 2
    for (int k = 0; k < K; k += 8) {
        const v2f a0 = *(const v2f*)(Ap + k);
        const v2f a1 = *(const v2f*)(Ap + k + 4);
        v2f b0, b1;
        if (TRANSB) {
            b0 = *(const v2f*)(Bp + k);
            b1 = *(const v2f*)(Bp + k + 4);
        } else {
            b0.x = Bp[(long)k * ldb];
            b0.y = Bp[(long)(k + 1) * ldb];
            b1.x = Bp[(long)(k + 4) * ldb];
            b1.y = Bp[(long)(k + 5) * ldb];
        }
        acc0 = __builtin_amdgcn_wmma_f32_16x16x4_f32(
            false, a0, false, b0, (short)0, acc0, false, false);
        acc1 = __builtin_amdgcn_wmma_f32_16x16x4_f32(
            false, a1, false, b1, (short)0, acc1, false, false);
    }
    const v8f acc = acc0 + acc1;

    // C/D layout: VGPR v -> row mt*16 + (lane>=16 ? 8 : 0) + v, col nt*16 + l
    const long crow0 = (long)mt * 16 + half * 8;
    const float bv = bias ? bias[bcol] : 0.0f;
#pragma unroll
    for (int v = 0; v < 8; ++v) {
        const long crow = crow0 + v;
        if (crow < (long)M) C[crow * (long)ldc + bcol] = acc[v] + bv;
    }
}

// ---------------------------------------------------------------------------
// Elementwise / scatter helpers
// ---------------------------------------------------------------------------
__global__ void fill_kernel(float* p, long n, float v) {
    long i = (long)blockIdx.x * blockDim.x + threadIdx.x;
    if (i < n) p[i] = v;
}

__global__ void scatter_count_kernel(float* deg, const int* dst, int e) {
    int i = blockIdx.x * blockDim.x + threadIdx.x;
    if (i < e) atomicAdd(&deg[dst[i]], 1.0f);
}

// RGCN edge scatter for one relation r: agg[dst] += hr[src] where etype == r.
// One thread handles 4 contiguous features (float4 load, 4 atomics).
__global__ void rgcn_scatter_kernel(float* __restrict__ agg,
                                    const float* __restrict__ hr,
                                    const int* __restrict__ src,
                                    const int* __restrict__ dst,
                                    const int* __restrict__ et,
                                    int r, int e) {
    long idx = (long)blockIdx.x * blockDim.x + threadIdx.x;
    int ed = (int)(idx >> 5);
    if (ed >= e) return;
    if (et[ed] != r) return;
    int q = (int)(idx & 31) * 4;
    const float4 v = *(const float4*)(hr + (long)src[ed] * DIM + q);
    float* p = agg + (long)dst[ed] * DIM + q;
    atomicAdd(p + 0, v.x);
    atomicAdd(p + 1, v.y);
    atomicAdd(p + 2, v.z);
    atomicAdd(p + 3, v.w);
}

// kg_h = relu(agg / max(deg,1) + selfh)
__global__ void rgcn_finish_kernel(float* __restrict__ out,
                                   const float* __restrict__ agg,
                                   const float* __restrict__ selfh,
                                   const float* __restrict__ deg, long n) {
    long i = (long)blockIdx.x * blockDim.x + threadIdx.x;
    if (i >= n) return;
    float d = fmaxf(deg[i >> 7], 1.0f);
    float v = agg[i] / d + selfh[i];
    out[i] = fmaxf(v, 0.0f);
}

// out = h / deg (per-row deg; 128 features)
__global__ void gcn_init_kernel(float* __restrict__ out,
                                const float* __restrict__ h,
                                const float* __restrict__ deg, long n) {
    long i = (long)blockIdx.x * blockDim.x + threadIdx.x;
    if (i >= n) return;
    out[i] = h[i] / deg[i >> 7];
}

// out[dst] += h[src] * deg[src]^-1/2 * deg[dst]^-1/2
__global__ void gcn_edge_kernel(float* __restrict__ out,
                                const float* __restrict__ h,
                                const int* __restrict__ src,
                                const int* __restrict__ dst,
                                const float* __restrict__ deg, int e) {
    long idx = (long)blockIdx.x * blockDim.x + threadIdx.x;
    int ed = (int)(idx >> 5);
    if (ed >= e) return;
    int q = (int)(idx & 31) * 4;
    int s = src[ed], t = dst[ed];
    float w = (1.0f / sqrtf(deg[s])) * (1.0f / sqrtf(deg[t]));
    const float4 v = *(const float4*)(h + (long)s * DIM + q);
    float* p = out + (long)t * DIM + q;
    atomicAdd(p + 0, v.x * w);
    atomicAdd(p + 1, v.y * w);
    atomicAdd(p + 2, v.z * w);
    atomicAdd(p + 3, v.w * w);
}

__global__ void relu_kernel(float* p, long n) {
    long i = (long)blockIdx.x * blockDim.x + threadIdx.x;
    if (i < n) p[i] = fmaxf(p[i], 0.0f);
}

__global__ void avg_kernel(float* __restrict__ out, const float* __restrict__ a,
                           const float* __restrict__ b, long n) {
    long i = (long)blockIdx.x * blockDim.x + threadIdx.x;
    if (i < n) out[i] = 0.5f * (a[i] + b[i]);
}

// x[row, :] /= max(rs[row], eps)   (128 features per row)
__global__ void rowdiv_kernel(float* __restrict__ x, const float* __restrict__ rs,
                              long n, float eps) {
    long i = (long)blockIdx.x * blockDim.x + threadIdx.x;
    if (i >= n) return;
    x[i] = x[i] / fmaxf(rs[i >> 7], eps);
}

// out[p, :] = leaky_relu(mat[index[p, col], :])
__global__ void gather_lrelu_kernel(float* __restrict__ out,
                                    const float* __restrict__ mat,
                                    const int* __restrict__ index,
                                    int col, long n) {
    long i = (long)blockIdx.x * blockDim.x + threadIdx.x;
    if (i >= n) return;
    long p = i >> 7;
    int d = (int)(i & 127);
    int row = index[p * 2 + col];
    float v = mat[(long)row * DIM + d];
    out[i] = (v >= 0.0f) ? v : 0.01f * v;
}

// ---------------------------------------------------------------------------
// Block-per-row reductions
// ---------------------------------------------------------------------------
__global__ void rowsum_kernel(const float* __restrict__ A, float* __restrict__ out,
                              int cols) {
    __shared__ float sm[256];
    const int r = blockIdx.x;
    float s = 0.0f;
    for (int c = threadIdx.x; c < cols; c += blockDim.x)
        s += A[(long)r * cols + c];
    sm[threadIdx.x] = s;
    __syncthreads();
    for (int st = 128; st > 0; st >>= 1) {
        if ((int)threadIdx.x < st) sm[threadIdx.x] += sm[threadIdx.x + st];
        __syncthreads();
    }
    if (threadIdx.x == 0) out[r] = sm[0];
}

// Y[row,:] = X[row,:] / max(||X[row,:]||2, 1e-12)   (Y may alias X)
__global__ void l2norm_rows_kernel(const float* __restrict__ X,
                                   float* __restrict__ Y, int cols) {
    __shared__ float sm[256];
    const int r = blockIdx.x;
    float ss = 0.0f;
    for (int c = threadIdx.x; c < cols; c += blockDim.x) {
        float v = X[(long)r * cols + c];
        ss += v * v;
    }
    sm[threadIdx.x] = ss;
    __syncthreads();
    for (int st = 128; st > 0; st >>= 1) {
        if ((int)threadIdx.x < st) sm[threadIdx.x] += sm[threadIdx.x + st];
        __syncthreads();
    }
    float inv = 1.0f / fmaxf(sqrtf(sm[0]), 1e-12f);
    __syncthreads();
    for (int c = threadIdx.x; c < cols; c += blockDim.x)
        Y[(long)r * cols + c] = X[(long)r * cols + c] * inv;
}

// Cross-entropy over rows of S (+1e-8 to logits): loss += -logp[r, tgt[r]] / rows
__global__ void ce_loss_kernel(const float* __restrict__ S,
                               const int* __restrict__ tgt,
                               float* __restrict__ loss, int cols, int rows) {
    __shared__ float sm[256];
    const int r = blockIdx.x;
    float mx = -1e30f;
    for (int c = threadIdx.x; c < cols; c += blockDim.x)
        mx = fmaxf(mx, S[(long)r * cols + c] + 1e-8f);
    sm[threadIdx.x] = mx;
    __syncthreads();
    for (int st = 128; st > 0; st >>= 1) {
        if ((int)threadIdx.x < st) sm[threadIdx.x] = fmaxf(sm[threadIdx.x], sm[threadIdx.x + st]);
        __syncthreads();
    }
    mx = sm[0];
    __syncthreads();
    float se = 0.0f;
    for (int c = threadIdx.x; c < cols; c += blockDim.x)
        se += expf(S[(long)r * cols + c] + 1e-8f - mx);
    sm[threadIdx.x] = se;
    __syncthreads();
    for (int st = 128; st > 0; st >>= 1) {
        if ((int)threadIdx.x < st) sm[threadIdx.x] += sm[threadIdx.x + st];
        __syncthreads();
    }
    if (threadIdx.x == 0) {
        float lt = S[(long)r * cols + tgt[r]] + 1e-8f;
        float lp = (lt - mx) - logf(sm[0]);
        atomicAdd(loss, -lp / (float)rows);
    }
}

// NCE over cosine-sim matrix S (rows = cols = n, TAU = 0.5):
// loss += lam * mean_r( -log( exp(S[r,r]/TAU) / sum_c exp(S[r,c]/TAU) + 1e-8 ) )
__global__ void nce_loss_kernel(const float* __restrict__ S, int n,
                                const float* __restrict__ lam,
                                float* __restrict__ loss) {
    __shared__ float sm[256];
    const int r = blockIdx.x;
    float se = 0.0f;
    for (int c = threadIdx.x; c < n; c += blockDim.x)
        se += expf(S[(long)r * n + c] * 2.0f);   // 1/TAU = 2
    sm[threadIdx.x] = se;
    __syncthreads();
    for (int st = 128; st > 0; st >>= 1) {
        if ((int)threadIdx.x < st) sm[threadIdx.x] += sm[threadIdx.x + st];
        __syncthreads();
    }
    if (threadIdx.x == 0) {
        float pos = expf(S[(long)r * n + r] * 2.0f);
        float li = -logf(pos / sm[0] + 1e-8f);
        atomicAdd(loss, lam[0] * li / (float)n);
    }
}

// ---------------------------------------------------------------------------
// Host-side launch helpers
// ---------------------------------------------------------------------------
static inline void launch_gemm(const float* A, const float* B, const float* bias,
                               float* C, int M, int N, int K,
                               int lda, int ldb, int ldc, int transB,
                               hipStream_t s) {
    const int tiles = ((M + 15) / 16) * (N / 16);
    dim3 g((tiles + 7) / 8, 1, 1);
    if (transB)
        wmma_gemm_f32<1><<<g, 256, 0, s>>>(A, B, bias, C, M, N, K, lda, ldb, ldc);
    else
        wmma_gemm_f32<0><<<g, 256, 0, s>>>(A, B, bias, C, M, N, K, lda, ldb, ldc);
}

static inline void launch_fill(float* p, long n, float v, hipStream_t s) {
    fill_kernel<<<(unsigned)((n + 255) / 256), 256, 0, s>>>(p, n, v);
}

extern "C" void kernel_launch(void* const* d_in, const int* in_sizes, int n_in,
                              void* d_out, int out_size, void* d_ws, size_t ws_size,
                              hipStream_t stream) {
    // ---- inputs (setup_inputs order) ----
    const float* ent_emb  = (const float*)d_in[0];   // [N_ENT, DIM]
    const float* W_rel    = (const float*)d_in[1];   // [R2, DIM, DIM]
    const float* W_self   = (const float*)d_in[2];   // [DIM, DIM]
    const float* gcn_W1   = (const float*)d_in[3];   // [DIM, DIM]
    const float* gcn_W2   = (const float*)d_in[4];   // [DIM, DIM]
    const float* proj1_W  = (const float*)d_in[5];   // [DIM, DIM]
    const float* proj1_b  = (const float*)d_in[6];   // [DIM]
    const float* proj2_W  = (const float*)d_in[7];   // [DIS, DIM]
    const float* proj2_b  = (const float*)d_in[8];   // [DIM]
    const float* g2o      = (const float*)d_in[9];   // [NG, N_ENT]
    const float* d2g      = (const float*)d_in[10];  // [ND, NG]
    const float* d_h2     = (const float*)d_in[11];  // [ND, DIS]
    const float* lam_1    = (const float*)d_in[12];
    const float* lam_2    = (const float*)d_in[13];
    const int*   kg_src   = (const int*)d_in[14];
    const int*   kg_dst   = (const int*)d_in[15];
    const int*   kg_type  = (const int*)d_in[16];
    const int*   g_src    = (const int*)d_in[17];
    const int*   g_dst    = (const int*)d_in[18];
    const int*   gsvd_src = (const int*)d_in[19];
    const int*   gsvd_dst = (const int*)d_in[20];
    const int*   index    = (const int*)d_in[21];    // [NP, 2]
    const int*   sim_true = (const int*)d_in[22];    // [NP]
    (void)in_sizes; (void)n_in; (void)out_size; (void)ws_size;

    // ---- outputs ----
    float* out       = (float*)d_out;
    float* d_z_out   = out;                               // NP*DIM
    float* d_emb_out = out + (long)NP * DIM;              // ND*DIM
    float* sim_out   = d_emb_out + (long)ND * DIM;        // NP*NP
    float* loss_out  = sim_out + (long)NP * NP;           // 1

    // ---- workspace carve (floats) ----
    float* w = (float*)d_ws;
    float* hr      = w; w += (long)N_ENT * DIM;   // per-relation transformed emb
    float* agg     = w; w += (long)N_ENT * DIM;
    float* selfh   = w; w += (long)N_ENT * DIM;
    float* kg_h    = w; w += (long)N_ENT * DIM;
    float* deg_kg  = w; w += N_ENT;
    float* rs_g2o  = w; w += NG;
    float* gx      = w; w += (long)NG * DIM;
    float* t0      = w; w += (long)NG * DIM;      // shared dense temp
    float* h1a     = w; w += (long)NG * DIM;
    float* h1b     = w; w += (long)NG * DIM;
    float* g_h1    = w; w += (long)NG * DIM;
    float* g_h2    = w; w += (long)NG * DIM;
    float* g_h     = w; w += (long)NG * DIM;
    float* deg_a   = w; w += NG;
    float* deg_b   = w; w += NG;
    float* rs_d2g  = w; w += ND;
    float* d_h1    = w; w += (long)ND * DIM;
    float* d_z2f   = w; w += (long)ND * DIM;
    float* d_z1    = w; w += (long)NP * DIM;
    float* d_z2    = w; w += (long)NP * DIM;
    float* xn1     = w; w += (long)NG * DIM;      // big enough for NP*DIM too
    float* xn2     = w; w += (long)NG * DIM;
    float* S       = w; w += (long)NG * NG;       // reused for 512x512 too

    const long ND_feat   = (long)N_ENT * DIM;
    const long NG_feat   = (long)NG * DIM;
    const long NDD_feat  = (long)ND * DIM;
    const long NP_feat   = (long)NP * DIM;

    // ================= RGCN =================
    launch_fill(deg_kg, N_ENT, 0.0f, stream);
    scatter_count_kernel<<<(EK + 255) / 256, 256, 0, stream>>>(deg_kg, kg_dst, EK);
    launch_fill(agg, ND_feat, 0.0f, stream);
    for (int r = 0; r < R2; ++r) {
        launch_gemm(ent_emb, W_rel + (long)r * DIM * DIM, nullptr, hr,
                    N_ENT, DIM, DIM, DIM, DIM, DIM, 0, stream);
        const long nth = (long)EK * 32;
        rgcn_scatter_kernel<<<(unsigned)((nth + 255) / 256), 256, 0, stream>>>(
            agg, hr, kg_src, kg_dst, kg_type, r, EK);
    }
    launch_gemm(ent_emb, W_self, nullptr, selfh, N_ENT, DIM, DIM, DIM, DIM, DIM, 0, stream);
    rgcn_finish_kernel<<<(unsigned)((ND_feat + 255) / 256), 256, 0, stream>>>(
        kg_h, agg, selfh, deg_kg, ND_feat);

    // ================= pool g2o =================
    rowsum_kernel<<<NG, 256, 0, stream>>>(g2o, rs_g2o, N_ENT);
    launch_gemm(g2o, kg_h, nullptr, gx, NG, DIM, N_ENT, N_ENT, DIM, DIM, 0, stream);
    rowdiv_kernel<<<(unsigned)((NG_feat + 255) / 256), 256, 0, stream>>>(gx, rs_g2o, NG_feat, 1e-8f);

    // ================= twin GCNs (shared layer-1 GEMM) =================
    launch_fill(deg_a, NG, 1.0f, stream);
    scatter_count_kernel<<<(EG + 255) / 256, 256, 0, stream>>>(deg_a, g_dst, EG);
    launch_fill(deg_b, NG, 1.0f, stream);
    scatter_count_kernel<<<(EG + 255) / 256, 256, 0, stream>>>(deg_b, gsvd_dst, EG);

    const long eg_th = (long)EG * 32;
    // layer 1: t0 = gx @ W1 (shared), aggregate per edge set, relu
    launch_gemm(gx, gcn_W1, nullptr, t0, NG, DIM, DIM, DIM, DIM, DIM, 0, stream);
    gcn_init_kernel<<<(unsigned)((NG_feat + 255) / 256), 256, 0, stream>>>(h1a, t0, deg_a, NG_feat);
    gcn_edge_kernel<<<(unsigned)((eg_th + 255) / 256), 256, 0, stream>>>(h1a, t0, g_src, g_dst, deg_a, EG);
    relu_kernel<<<(unsigned)((NG_feat + 255) / 256), 256, 0, stream>>>(h1a, NG_feat);
    gcn_init_kernel<<<(unsigned)((NG_feat + 255) / 256), 256, 0, stream>>>(h1b, t0, deg_b, NG_feat);
    gcn_edge_kernel<<<(unsigned)((eg_th + 255) / 256), 256, 0, stream>>>(h1b, t0, gsvd_src, gsvd_dst, deg_b, EG);
    relu_kernel<<<(unsigned)((NG_feat + 255) / 256), 256, 0, stream>>>(h1b, NG_feat);
    // layer 2, graph A
    launch_gemm(h1a, gcn_W2, nullptr, t0, NG, DIM, DIM, DIM, DIM, DIM, 0, stream);
    gcn_init_kernel<<<(unsigned)((NG_feat + 255) / 256), 256, 0, stream>>>(g_h1, t0, deg_a, NG_feat);
    gcn_edge_kernel<<<(unsigned)((eg_th + 255) / 256), 256, 0, stream>>>(g_h1, t0, g_src, g_dst, deg_a, EG);
    // layer 2, graph B
    launch_gemm(h1b, gcn_W2, nullptr, t0, NG, DIM, DIM, DIM, DIM, DIM, 0, stream);
    gcn_init_kernel<<<(unsigned)((NG_feat + 255) / 256), 256, 0, stream>>>(g_h2, t0, deg_b, NG_feat);
    gcn_edge_kernel<<<(unsigned)((eg_th + 255) / 256), 256, 0, stream>>>(g_h2, t0, gsvd_src, gsvd_dst, deg_b, EG);

    avg_kernel<<<(unsigned)((NG_feat + 255) / 256), 256, 0, stream>>>(g_h, g_h1, g_h2, NG_feat);

    // ================= pool d2g + projections =================
    rowsum_kernel<<<ND, 256, 0, stream>>>(d2g, rs_d2g, NG);
    launch_gemm(d2g, g_h, nullptr, d_h1, ND, DIM, NG, NG, DIM, DIM, 0, stream);
    rowdiv_kernel<<<(unsigned)((NDD_feat + 255) / 256), 256, 0, stream>>>(d_h1, rs_d2g, NDD_feat, 1e-8f);

    launch_gemm(d_h1, proj1_W, proj1_b, d_emb_out, ND, DIM, DIM, DIM, DIM, DIM, 0, stream);
    launch_gemm(d_h2, proj2_W, proj2_b, d_z2f, ND, DIM, DIS, DIS, DIM, DIM, 0, stream);

    // ================= pair gather + d_z =================
    gather_lrelu_kernel<<<(unsigned)((NP_feat + 255) / 256), 256, 0, stream>>>(d_z1, d_emb_out, index, 0, NP_feat);
    gather_lrelu_kernel<<<(unsigned)((NP_feat + 255) / 256), 256, 0, stream>>>(d_z2, d_z2f, index, 1, NP_feat);
    avg_kernel<<<(unsigned)((NP_feat + 255) / 256), 256, 0, stream>>>(d_z_out, d_z1, d_z2, NP_feat);

    // ================= similarity + losses =================
    launch_fill(loss_out, 1, 0.0f, stream);

    // similarity = l2norm_rows(d_z @ d_z^T)
    launch_gemm(d_z_out, d_z_out, nullptr, sim_out, NP, NP, DIM, DIM, DIM, NP, 1, stream);
    l2norm_rows_kernel<<<NP, 256, 0, stream>>>(sim_out, sim_out, NP);
    ce_loss_kernel<<<NP, 256, 0, stream>>>(sim_out, sim_true, loss_out, NP, NP);

    // NCE over graphs (2000x2000)
    l2norm_rows_kernel<<<NG, 256, 0, stream>>>(g_h1, xn1, DIM);
    l2norm_rows_kernel<<<NG, 256, 0, stream>>>(g_h2, xn2, DIM);
    launch_gemm(xn1, xn2, nullptr, S, NG, NG, DIM, DIM, DIM, NG, 1, stream);
    nce_loss_kernel<<<NG, 256, 0, stream>>>(S, NG, lam_1, loss_out);

    // NCE over diseases (512x512), reuse xn1/xn2/S
    l2norm_rows_kernel<<<NP, 256, 0, stream>>>(d_z1, xn1, DIM);
    l2norm_rows_kernel<<<NP, 256, 0, stream>>>(d_z2, xn2, DIM);
    launch_gemm(xn1, xn2, nullptr, S, NP, NP, DIM, DIM, DIM, NP, 1, stream);
    nce_loss_kernel<<<NP, 256, 0, stream>>>(S, NP, lam_2, loss_out);
}